// ExactTopKAttention_47304769798226
// MI455X (gfx1250) — compile-verified
//
#include <hip/hip_runtime.h>
#include <hip/hip_bf16.h>

// ---- problem constants (from reference setup_inputs) ----
constexpr int T = 2048, S = 2048, H = 16, E = 64, TOPK = 32;
constexpr int QTILE   = 16;            // queries per workgroup (WMMA M)
constexpr int NWAVES  = 8;             // 256 threads, wave32
constexpr int SSTRIDE = 2052;          // 2048 + 4 pad -> halves land 32 banks apart
constexpr float SM_TEMP = 0.125f;      // 1/sqrt(64)

typedef float v2f __attribute__((ext_vector_type(2)));
typedef float v8f __attribute__((ext_vector_type(8)));

__global__ __launch_bounds__(256) void
ExactTopKAttention_47304769798226_kernel(const float* __restrict__ Qg,
                                         const float* __restrict__ Kg,
                                         const float* __restrict__ Vg,
                                         float* __restrict__ Og)
{
    extern __shared__ float smem[];
    float* sc   = smem;                         // [QTILE][SSTRIDE] score panel
    float* topv = smem + QTILE * SSTRIDE;       // [QTILE][TOPK] kept scores (desc)
    int*   topc = (int*)(topv + QTILE * TOPK);  // [QTILE][TOPK] kept key indices

    const int h    = blockIdx.x % H;
    const int t0   = (blockIdx.x / H) * QTILE;
    const int lane = threadIdx.x & 31;
    const int wave = threadIdx.x >> 5;
    const int n    = lane & 15;   // M (for A) / N (for B,C) index
    const int kh   = lane >> 4;   // K-half selector for 16x16x4 operand layout

    // ================= Phase 1: scores = (Q*temp) @ K^T via WMMA f32 =========
    // A-operand (16x4 f32): lanes 0-15 hold M=lane {K,K+1}; lanes 16-31 {K+2,K+3}
    v2f qa[16];
    {
        const float* qrow = Qg + ((size_t)(t0 + n) * H + h) * E;
        #pragma unroll
        for (int kk = 0; kk < 16; ++kk) {
            v2f q = *(const v2f*)(qrow + 4 * kk + 2 * kh);
            qa[kk] = q * SM_TEMP;               // fold softmax temperature into Q
        }
    }

    for (int tile = wave; tile < S / 16; tile += NWAVES) {
        const int s0 = tile * 16;
        const float* krow = Kg + ((size_t)(s0 + n) * H + h) * E;
        // prefetch next key tile this wave will touch (global_prefetch_b8)
        if (tile + NWAVES < S / 16)
            __builtin_prefetch(Kg + ((size_t)(s0 + NWAVES * 16 + n) * H + h) * E, 0, 3);

        // two independent accumulator chains -> ILP on the matrix pipe
        v8f c0 = {};
        v8f c1 = {};
        #pragma unroll
        for (int kk = 0; kk < 8; ++kk) {
            v2f b0 = *(const v2f*)(krow + 4 * (2 * kk)     + 2 * kh);
            v2f b1 = *(const v2f*)(krow + 4 * (2 * kk + 1) + 2 * kh);
            c0 = __builtin_amdgcn_wmma_f32_16x16x4_f32(
                     false, qa[2 * kk],     false, b0, (short)0, c0, false, false);
            c1 = __builtin_amdgcn_wmma_f32_16x16x4_f32(
                     false, qa[2 * kk + 1], false, b1, (short)0, c1, false, false);
        }
        v8f c = c0 + c1;
        // C layout: VGPR r -> row r (lanes 0-15) / row r+8 (lanes 16-31), col = s0+n
        #pragma unroll
        for (int r = 0; r < 8; ++r)
            sc[(r + 8 * kh) * SSTRIDE + s0 + n] = c[r];
    }
    __syncthreads();

    // ============ Phase 2: exact top-32, softmax, sparse AV (2 rows/wave) =====
    for (int rr = 0; rr < 2; ++rr) {
        const int row = wave * 2 + rr;

        // each lane caches its strided 64-element slice of the row in registers
        float val[64];
        #pragma unroll
        for (int i = 0; i < 64; ++i)
            val[i] = sc[row * SSTRIDE + lane + 32 * i];

        // per-lane running local max (only the winner rescans each round)
        float m = -INFINITY; int mi = 0;
        #pragma unroll
        for (int i = 0; i < 64; ++i) {
            bool gt = val[i] > m;
            m  = gt ? val[i] : m;
            mi = gt ? i : mi;
        }

        for (int t = 0; t < TOPK; ++t) {
            float g = m;
            #pragma unroll
            for (int off = 16; off >= 1; off >>= 1)
                g = fmaxf(g, __shfl_xor(g, off, 32));
            unsigned long long ball = __ballot(m == g);
            int src = (int)__ffsll(ball) - 1;
            if (lane == src) {
                topv[row * TOPK + t] = g;
                topc[row * TOPK + t] = lane + 32 * mi;
                #pragma unroll
                for (int i = 0; i < 64; ++i)           // retire without dyn index
                    val[i] = (i == mi) ? -INFINITY : val[i];
                m = -INFINITY; mi = 0;                 // rescan this lane only
                #pragma unroll
                for (int i = 0; i < 64; ++i) {
                    bool gt = val[i] > m;
                    m  = gt ? val[i] : m;
                    mi = gt ? i : mi;
                }
            }
        }

        // softmax over the 32 kept scores: lane i owns entry i (desc order)
        float kv = topv[row * TOPK + lane];
        int   kc = topc[row * TOPK + lane];
        float mx = __shfl(kv, 0, 32);                  // entry 0 is the max
        float e  = __expf(kv - mx);
        float ss = e;
        #pragma unroll
        for (int off = 16; off >= 1; off >>= 1)
            ss += __shfl_xor(ss, off, 32);
        float p = e / ss;

        // sparse AV: lane -> output dims {lane, lane+32}
        float acc0 = 0.f, acc1 = 0.f;
        #pragma unroll 4
        for (int i = 0; i < TOPK; ++i) {
            float pi = __shfl(p,  i, 32);
            int   ci = __shfl(kc, i, 32);
            const float* vrow = Vg + ((size_t)ci * H + h) * E;
            acc0 = fmaf(pi, vrow[lane],      acc0);
            acc1 = fmaf(pi, vrow[lane + 32], acc1);
        }
        float* orow = Og + ((size_t)(t0 + row) * H + h) * E;
        orow[lane]      = acc0;
        orow[lane + 32] = acc1;
    }
}

extern "C" void kernel_launch(void* const* d_in, const int* in_sizes, int n_in,
                              void* d_out, int out_size, void* d_ws, size_t ws_size,
                              hipStream_t stream) {
    (void)in_sizes; (void)n_in; (void)out_size; (void)d_ws; (void)ws_size;
    const float* Q = (const float*)d_in[0];
    const float* K = (const float*)d_in[1];
    const float* V = (const float*)d_in[2];
    float* O = (float*)d_out;

    const int nblocks = (T / QTILE) * H;                       // 2048
    const size_t smem = (size_t)(QTILE * SSTRIDE               // score panel
                                 + QTILE * TOPK) * sizeof(float)
                        + (size_t)(QTILE * TOPK) * sizeof(int);
    ExactTopKAttention_47304769798226_kernel<<<dim3(nblocks), dim3(256), smem, stream>>>(
        Q, K, V, O);
}